// DetectionConfidenceMap2keypoint_54752243089716
// MI455X (gfx1250) — compile-verified
//
#include <hip/hip_runtime.h>
#include <hip/hip_bf16.h>

// DetectionConfidenceMap2keypoint for MI455X (gfx1250, wave32).
// B=32, K=64, H=W=128, fp32 in/out.
//
// Outputs (concatenated flat in d_out):
//   [0, B*K*H*W)                  map_val_all  (channel softmax)
//   [B*K*H*W, +B*K*2)             keypoint     (round(kp_x/zeta), round(kp_y/zeta))
//   [B*K*H*W+B*K*2, +B*K)         get_zeta
//
// Workspace: kpx[B*K], kpy[B*K] partial sums (fp32).
//
// Cache policy (192 MB L2): K1 reads its 128 MB input non-temporally (read
// once, don't retain) and writes the 128 MB map with default RT so it lands
// in L2; K2 then re-reads the map (non-temporal, last use) mostly from L2,
// keeping HBM traffic near the 256 MB streaming minimum (~11 us @ 23.3 TB/s).

typedef float v2f __attribute__((ext_vector_type(2)));
typedef float v8f __attribute__((ext_vector_type(8)));

#define B_  32
#define K_  64
#define H_  128
#define W_  128
#define HW_ (H_ * W_)          // 16384
#define BK_ (B_ * K_)          // 2048

// ---------------------------------------------------------------------------
// Kernel 1: softmax over the channel dim (K=64) for each (b,h,w) pixel.
// Thread = one pixel. For a fixed k, a wave's 32 lanes touch 32 consecutive
// floats (128B) -> fully coalesced despite the 64KB k-stride.
// ---------------------------------------------------------------------------
__global__ __launch_bounds__(256) void softmax_channel_kernel(
    const float* __restrict__ in, float* __restrict__ out)
{
    const int p  = blockIdx.x * 256 + threadIdx.x;   // pixel id in [0, B*H*W)
    const int b  = p >> 14;                          // / HW_
    const int hw = p & (HW_ - 1);
    const long base = (long)b * (K_ * HW_) + hw;

    float x[K_];
#pragma unroll
    for (int k = 0; k < K_; ++k)
        x[k] = __builtin_nontemporal_load(in + base + (long)k * HW_);  // TH=NT

    float m = x[0];
#pragma unroll
    for (int k = 1; k < K_; ++k) m = fmaxf(m, x[k]);

    float s = 0.0f;
#pragma unroll
    for (int k = 0; k < K_; ++k) { x[k] = __expf(x[k] - m); s += x[k]; }

    const float inv = 1.0f / s;
#pragma unroll
    for (int k = 0; k < K_; ++k) out[base + (long)k * HW_] = x[k] * inv;
}

// ---------------------------------------------------------------------------
// Kernel 2: per-(b,k) spatial reductions with V_WMMA_F32_16X16X4_F32.
// One 256-thread block (8 waves) per 128x128 plane; wave w owns row stripe
// [16w, 16w+16). Two independent WMMA accumulator chains per wave (even/odd
// column chunks) to halve the D->C dependency depth. LDS combine at the end.
//
//   D(16x16) = A(16x4) x B(4x16) + C, fp32 exact.
//   A = 16-row x 4-col tile of the plane.
//   B col 0 = ones          -> D[m][0] accumulates plain row sums
//   B col 1 = column index  -> D[m][1] accumulates j-weighted row sums
//
// VGPR layouts per CDNA5 ISA 7.12.2:
//   A (16x4 f32, 2 VGPRs): lanes 0-15 rows M=0..15 {K=0 in V0, K=1 in V1},
//                          lanes 16-31 rows M=0..15 {K=2 in V0, K=3 in V1}.
//   B ( 4x16 f32, 2 VGPRs): lanes 0-15 col N=lane {K=0 in V0, K=1 in V1},
//                          lanes 16-31 col N=lane-16 {K=2 in V0, K=3 in V1}.
//   D (16x16 f32, 8 VGPRs): lanes 0-15 col N=lane, V[v] = row M=v;
//                          lanes 16-31 col N=lane-16, V[v] = row M=v+8.
// ---------------------------------------------------------------------------
__global__ __launch_bounds__(256) void reduce_wmma_kernel(
    const float* __restrict__ map,
    float* __restrict__ zeta_out,     // [B*K]  -> goes directly to d_out zeta
    float* __restrict__ kpx_ws,       // [B*K]
    float* __restrict__ kpy_ws)       // [B*K]
{
    const int  plane = blockIdx.x;                   // b*K + k
    const float* __restrict__ M = map + (long)plane * HW_;

    const int tid  = threadIdx.x;
    const int wave = tid >> 5;                       // 0..7 -> row stripe
    const int lane = tid & 31;                       // full wave, EXEC all 1s
    const int n    = lane & 15;
    const int hi   = lane >> 4;                      // 0: K pair {0,1}; 1: {2,3}
    const int r0   = wave * 16;
    const int kk0  = 2 * hi;

    const float* __restrict__ rowPtr = M + (r0 + n) * W_;  // A row of this lane

    // Precompute all 32 B-matrix chunks (this lane's 2 elements each) and pin
    // them in VGPRs with an empty-asm barrier so LLVM cannot rematerialize
    // them inside the loop (it otherwise rebuilds each chunk per WMMA with
    // cndmask/cvt chains). Steady-state loop becomes load_b64 + wmma only.
    float bbx[32], bby[32];
#pragma unroll
    for (int c = 0; c < 32; ++c) {
        const int cc = 4 * c;
        bbx[c] = (n == 0) ? 1.0f : ((n == 1) ? (float)(cc + kk0)     : 0.0f);
        bby[c] = (n == 0) ? 1.0f : ((n == 1) ? (float)(cc + kk0 + 1) : 0.0f);
        asm volatile("" : "+v"(bbx[c]), "+v"(bby[c]));   // opaque: no remat
    }

    v8f c0 = {0.f, 0.f, 0.f, 0.f, 0.f, 0.f, 0.f, 0.f};
    v8f c1 = {0.f, 0.f, 0.f, 0.f, 0.f, 0.f, 0.f, 0.f};

#pragma unroll
    for (int c = 0; c < 32; c += 2) {
        // even chunk -> accumulator c0
        {
            const int col = 4 * c + kk0;             // even -> 8B aligned
            const v2f a = __builtin_nontemporal_load((const v2f*)(rowPtr + col));
            v2f bb;  bb.x = bbx[c];  bb.y = bby[c];
            c0 = __builtin_amdgcn_wmma_f32_16x16x4_f32(
                     false, a, false, bb, (short)0, c0, false, false);
        }
        // odd chunk -> accumulator c1
        {
            const int col = 4 * (c + 1) + kk0;
            const v2f a = __builtin_nontemporal_load((const v2f*)(rowPtr + col));
            v2f bb;  bb.x = bbx[c + 1];  bb.y = bby[c + 1];
            c1 = __builtin_amdgcn_wmma_f32_16x16x4_f32(
                     false, a, false, bb, (short)0, c1, false, false);
        }
    }

    // d = D[m][nD] with m = v + 8*hi, nD = lane&15 (== n).
    float s0 = 0.0f, s1 = 0.0f;
#pragma unroll
    for (int v = 0; v < 8; ++v) {
        const float d = c0[v] + c1[v];
        s0 += d;                                  // sum_m D[m][n]
        s1 += (float)(r0 + 8 * hi + v) * d;       // sum_m row_idx * D[m][n]
    }

    // Column 0 of D lives in lanes 0 & 16; column 1 in lanes 1 & 17.
    const float z  = __shfl(s0, 0, 32) + __shfl(s0, 16, 32); // zeta partial
    const float sx = __shfl(s0, 1, 32) + __shfl(s0, 17, 32); // sum j * map
    const float sy = __shfl(s1, 0, 32) + __shfl(s1, 16, 32); // sum i * map

    __shared__ float zs[8], xs[8], ys[8];
    if (lane == 0) { zs[wave] = z; xs[wave] = sx; ys[wave] = sy; }
    __syncthreads();

    if (tid == 0) {
        float zt = 0.f, xt = 0.f, yt = 0.f;
#pragma unroll
        for (int w = 0; w < 8; ++w) { zt += zs[w]; xt += xs[w]; yt += ys[w]; }
        zeta_out[plane] = zt;
        kpx_ws[plane]   = xt;
        kpy_ws[plane]   = yt;
    }
}

// ---------------------------------------------------------------------------
// Kernel 3: keypoint[b,k] = (rint(kpx/zeta), rint(kpy/zeta))
// rintf = round-half-to-even, matching jnp.round.
// ---------------------------------------------------------------------------
__global__ void finalize_kernel(const float* __restrict__ zeta,
                                const float* __restrict__ kpx,
                                const float* __restrict__ kpy,
                                float* __restrict__ kp_out)
{
    const int i = blockIdx.x * blockDim.x + threadIdx.x;
    if (i < BK_) {
        const float z = zeta[i];
        kp_out[2 * i + 0] = rintf(kpx[i] / z);
        kp_out[2 * i + 1] = rintf(kpy[i] / z);
    }
}

extern "C" void kernel_launch(void* const* d_in, const int* in_sizes, int n_in,
                              void* d_out, int out_size, void* d_ws, size_t ws_size,
                              hipStream_t stream)
{
    const float* in  = (const float*)d_in[0];        // [B,K,H,W] fp32
    float* out  = (float*)d_out;
    float* map  = out;                               // B*K*H*W
    float* kp   = out + (size_t)B_ * K_ * HW_;       // B*K*2
    float* zeta = kp + (size_t)BK_ * 2;              // B*K

    float* kpx = (float*)d_ws;                       // B*K floats
    float* kpy = kpx + BK_;                          // B*K floats

    // 1) channel softmax: 524288 pixels / 256 = 2048 blocks
    softmax_channel_kernel<<<(B_ * HW_) / 256, 256, 0, stream>>>(in, map);

    // 2) per-plane spatial reductions via fp32 WMMA: 8 waves per (b,k)
    reduce_wmma_kernel<<<BK_, 256, 0, stream>>>(map, zeta, kpx, kpy);

    // 3) keypoints
    finalize_kernel<<<(BK_ + 255) / 256, 256, 0, stream>>>(zeta, kpx, kpy, kp);
}